// BinaryTree_18734647345500
// MI455X (gfx1250) — compile-verified
//
#include <hip/hip_runtime.h>

// f32 WMMA operand vectors for the 16x16x4 shape (wave32):
//   A: 16x4  f32 = 64 elems / 32 lanes = 2 VGPRs
//   B:  4x16 f32 = 64 elems / 32 lanes = 2 VGPRs
//   C/D: 16x16 f32 = 256 elems / 32 lanes = 8 VGPRs
typedef float v2f __attribute__((ext_vector_type(2)));
typedef float v8f __attribute__((ext_vector_type(8)));

#define TREE_DEPTH 20
#define NDIMS 128

__global__ __launch_bounds__(32)
void BinaryTree_path_prob_wmma(const float* __restrict__ W,
                               const int* __restrict__ vj_p,
                               const int* __restrict__ uk_p,
                               float* __restrict__ out)
{
    __shared__ float lds_x[NDIMS];   // the leaf row x = W[v_j + leaf_off]
    __shared__ float lds_d[512];     // two 16x16 f32 D tiles spilled for gather

    const int lane = threadIdx.x & 31;   // single wave32
    const int v = vj_p[0];
    const int u = uk_p[0];

    // ---- stage x into LDS: 32 lanes x float4 = 128 floats, one b128 each ----
    const int leaf_off = (1 << TREE_DEPTH) - 1;
    const float4* xrow = (const float4*)(W + (size_t)(v + leaf_off) * NDIMS);
    ((float4*)lds_x)[lane] = xrow[lane];
    __syncthreads();

    // ---- per-lane A-row (path node) indices ----
    // path node at level l: ((u + 2^20) >> (20 - l)) - 1, l = 0..20
    const int m    = lane & 15;      // M within the 16-row tile
    const int hi   = lane >> 4;      // 0: K=0,1 half  1: K=2,3 half
    const int koff = hi * 2;

    const int t    = u + (1 << TREE_DEPTH);
    const int lvl1 = m;              // tile 1: levels 0..15
    const int lvl2 = 16 + m;         // tile 2: levels 16..20 valid, rest padded
    const int row1 = (t >> (TREE_DEPTH - lvl1)) - 1;
    const int row2 = (lvl2 <= TREE_DEPTH) ? ((t >> (TREE_DEPTH - lvl2)) - 1) : 0;

    const float* __restrict__ r1 = W + (size_t)row1 * NDIMS + koff;
    const float* __restrict__ r2 = W + (size_t)row2 * NDIMS + koff;

    // B matrix 4x16: only column N=0 is non-zero (holds x). Lanes 0 and 16
    // carry it (N = lane % 16, K pair selected by lane >> 4).
    const bool bcol = (m == 0);

    v8f c1 = {0.f,0.f,0.f,0.f,0.f,0.f,0.f,0.f};
    v8f c2 = {0.f,0.f,0.f,0.f,0.f,0.f,0.f,0.f};

    // ---- K loop: 128 = 32 chunks of K=4; two WMMA tiles per chunk ----
    #pragma unroll
    for (int k = 0; k < NDIMS / 4; ++k) {
        const int kb = k * 4;
        v2f a1 = *(const v2f*)(r1 + kb);            // global_load_b64
        v2f a2 = *(const v2f*)(r2 + kb);            // global_load_b64
        v2f xb = *(const v2f*)(lds_x + kb + koff);  // ds_load_b64
        v2f b;
        b.x = bcol ? xb.x : 0.0f;
        b.y = bcol ? xb.y : 0.0f;
        // 8 args: (neg_a, A, neg_b, B, c_mod, C, reuse_a, reuse_b)
        c1 = __builtin_amdgcn_wmma_f32_16x16x4_f32(false, a1, false, b,
                                                   (short)0, c1, false, false);
        c2 = __builtin_amdgcn_wmma_f32_16x16x4_f32(false, a2, false, b,
                                                   (short)0, c2, false, false);
    }

    // ---- spill D tiles: lane L's 8 accumulator VGPRs -> lds_d[L*8 + r] ----
    #pragma unroll
    for (int r = 0; r < 8; ++r) {
        lds_d[lane * 8 + r]       = c1[r];
        lds_d[256 + lane * 8 + r] = c2[r];
    }
    __syncthreads();

    // ---- lane 0: gather the 21 logits from column N=0, reduce ----
    // D[m][0] location: m<8 -> lane 0, VGPR m ; m>=8 -> lane 16, VGPR m-8.
    if (lane == 0) {
        float prob = 1.0f;
        #pragma unroll
        for (int l = 0; l <= TREE_DEPTH; ++l) {
            float logit;
            if (l < 8) {
                logit = lds_d[l];                    // tile1, lane 0, vgpr l
            } else if (l < 16) {
                logit = lds_d[16 * 8 + (l - 8)];     // tile1, lane 16, vgpr l-8
            } else {
                logit = lds_d[256 + (l - 16)];       // tile2, lane 0, vgpr l-16
            }
            prob *= 1.0f / (1.0f + __expf(-logit));  // sigmoid
        }
        out[0] = prob;                               // [1,1] scalar
    }
}

extern "C" void kernel_launch(void* const* d_in, const int* in_sizes, int n_in,
                              void* d_out, int out_size, void* d_ws, size_t ws_size,
                              hipStream_t stream)
{
    (void)in_sizes; (void)n_in; (void)out_size; (void)d_ws; (void)ws_size;
    const float* W  = (const float*)d_in[0];   // [SIZE, 128] f32
    const int*   vj = (const int*)d_in[1];     // scalar v_j_idx
    const int*   uk = (const int*)d_in[2];     // scalar u_k_idx
    float* out = (float*)d_out;                // 1 float

    // Whole problem (~11 KB reads, ~5 KFLOP) fits one wave32 on one WGP:
    // latency-bound, so a single block of 32 threads is the right shape.
    BinaryTree_path_prob_wmma<<<1, 32, 0, stream>>>(W, vj, uk, out);
}